// FDSCS_46583215292578
// MI455X (gfx1250) — compile-verified
//
#include <hip/hip_runtime.h>
#include <hip/hip_bf16.h>
#include <stdint.h>

// ---------------- problem dimensions (fixed by the reference) ----------------
static constexpr int NB   = 2;     // batch
static constexpr int HIN  = 384, WIN = 1280;
static constexpr int H    = 192, W   = 640;   // pooled
static constexpr int D    = 128;              // MAXDISP
static constexpr int NHW  = NB * H * W;       // 245760 elems per plane

// normalization: (c - mean)/std  ==  fma(c, A, B)
static constexpr float AY = 1.0f / 0.1949711f;
static constexpr float BY = -11.08282948f / 0.1949711f;
static constexpr float AU = 1.0f / 35.91432953f;
static constexpr float BU = -0.02175535f / 35.91432953f;
static constexpr float AV = 1.0f / 26.79782867f;
static constexpr float BV = -0.02679042f / 26.79782867f;

typedef __attribute__((ext_vector_type(4))) float v4f;
typedef __attribute__((ext_vector_type(2))) float v2f;
typedef __attribute__((ext_vector_type(4))) int   v4i;

// address-space-qualified vector pointers for the async builtin
typedef __attribute__((address_space(1))) v4i g_v4i;   // global
typedef __attribute__((address_space(3))) v4i l_v4i;   // LDS

// ---------------- CDNA5 async global->LDS staging ----------------
#if defined(__has_builtin)
#if __has_builtin(__builtin_amdgcn_global_load_async_to_lds_b128) && \
    __has_builtin(__builtin_amdgcn_s_wait_asynccnt)
#define FDSCS_ASYNC 1
#endif
#endif
#ifndef FDSCS_ASYNC
#define FDSCS_ASYNC 0
#warning "gfx1250 async-to-LDS builtins unavailable; falling back to synchronous staging"
#endif

__device__ __forceinline__ void stage16(const void* g, void* l) {
#if FDSCS_ASYNC
  __builtin_amdgcn_global_load_async_to_lds_b128(
      (g_v4i*)g, (l_v4i*)l, /*offset=*/0, /*cpol=*/0);
#else
  *(v4i*)l = *(const v4i*)g;
#endif
}

__device__ __forceinline__ void stage_wait() {
#if FDSCS_ASYNC
  __builtin_amdgcn_s_wait_asynccnt(0);
#endif
}

// ---------------- kernel 1: 2x2 avg pool + RGB->YCbCr ----------------
// grid = (W/128, H, NB), block = 64; each thread -> 2 adjacent output pixels,
// so every global load is a coalesced 16-byte b128.
__global__ __launch_bounds__(64) void fdscs_pool_ycbcr(
    const float* __restrict__ img,   // (NB,3,HIN,WIN)
    float* __restrict__ Yp, float* __restrict__ CBp, float* __restrict__ CRp) {
  const int xo = (blockIdx.x * 64 + threadIdx.x) * 2;   // even, < W
  const int y  = blockIdx.y;
  const int n  = blockIdx.z;

  const float* p = img + (size_t)n * 3 * HIN * WIN + (size_t)(2 * y) * WIN + 2 * xo;
  float c0[3], c1[3];
#pragma unroll
  for (int c = 0; c < 3; ++c) {
    const float* pc = p + (size_t)c * HIN * WIN;
    v4f t = *(const v4f*)pc;
    v4f b = *(const v4f*)(pc + WIN);
    c0[c] = 0.25f * (t.x + t.y + b.x + b.y);
    c1[c] = 0.25f * (t.z + t.w + b.z + b.w);
  }
  float y0 = 0.299f * c0[0] + 0.587f * c0[1] + 0.114f * c0[2];
  float y1 = 0.299f * c1[0] + 0.587f * c1[1] + 0.114f * c1[2];
  v2f yv  = {y0, y1};
  v2f cbv = {(c0[2] - y0) * 0.564f + 0.5f, (c1[2] - y1) * 0.564f + 0.5f};
  v2f crv = {(c0[0] - y0) * 0.713f + 0.5f, (c1[0] - y1) * 0.713f + 0.5f};

  const int idx = (n * H + y) * W + xo;
  *(v2f*)&Yp[idx]  = yv;
  *(v2f*)&CBp[idx] = cbv;
  *(v2f*)&CRp[idx] = crv;
}

// ---------------- kernel 2: 5x5 census (24 bits, MSB-first, center skipped) --
// grid = (W/128, H, NB), block = 128. All reads hit L2 (planes are ~1 MB).
__global__ __launch_bounds__(128) void fdscs_census(
    const float* __restrict__ Yp, int* __restrict__ CEN) {
  const int x = blockIdx.x * 128 + threadIdx.x;
  const int y = blockIdx.y;
  const int n = blockIdx.z;
  int code = 0;
  if (x >= 2 && x < W - 2 && y >= 2 && y < H - 2) {
    const float* base = Yp + (size_t)n * H * W;
    float c = base[y * W + x];
    int k = 0;
#pragma unroll
    for (int v = 0; v < 5; ++v) {
#pragma unroll
      for (int u = 0; u < 5; ++u) {
        if (u == 2 && v == 2) continue;
        float nb = base[(y - 2 + v) * W + (x - 2 + u)];
        code |= (nb >= c ? 1 : 0) << (23 - k);
        ++k;
      }
    }
  }
  CEN[(n * H + y) * W + x] = code;
}

// ---------------- kernel 3: fused 3-channel cost volume + normalize ----------
// grid = (H, NB, D/DCHUNK), block = 160 threads (5 waves), thread owns 4 x's.
static constexpr int DCHUNK = 16;

__global__ __launch_bounds__(160) void fdscs_costvol(
    const int* __restrict__ lcen, const float* __restrict__ lcb, const float* __restrict__ lcr,
    const int* __restrict__ rcen, const float* __restrict__ rcb, const float* __restrict__ rcr,
    float* __restrict__ out) {
  __shared__ __align__(16) int   s_lcen[W];
  __shared__ __align__(16) float s_lcb[W];
  __shared__ __align__(16) float s_lcr[W];
  __shared__ __align__(16) int   s_rcen[W];
  __shared__ __align__(16) float s_rcb[W];
  __shared__ __align__(16) float s_rcr[W];

  const int x0 = threadIdx.x * 4;
  const int y  = blockIdx.x;
  const int n  = blockIdx.y;
  const int d0 = blockIdx.z * DCHUNK;
  const int rowoff = (n * H + y) * W + x0;

  // async-stage the full row (left + right, 3 planes each) into LDS
  stage16(lcen + rowoff, &s_lcen[x0]);
  stage16(lcb  + rowoff, &s_lcb[x0]);
  stage16(lcr  + rowoff, &s_lcr[x0]);
  stage16(rcen + rowoff, &s_rcen[x0]);
  stage16(rcb  + rowoff, &s_rcb[x0]);
  stage16(rcr  + rowoff, &s_rcr[x0]);
  stage_wait();
  __syncthreads();

  // right-image values for this thread's quad are disparity-invariant: hoist.
  const v4i rc  = *(const v4i*)&s_rcen[x0];
  const v4f rcb4 = *(const v4f*)&s_rcb[x0];
  const v4f rcr4 = *(const v4f*)&s_rcr[x0];

  const size_t chs = (size_t)H * W;                       // per-channel plane
  const size_t ob  = (size_t)n * 384 * chs + (size_t)y * W + x0;

  for (int dd = 0; dd < DCHUNK; ++dd) {
    const int d = d0 + dd;
    v4f oh, ou, ov;
#pragma unroll
    for (int j = 0; j < 4; ++j) {
      const int xl = x0 + j + d;                          // left sample
      float h = 0.0f, u = 0.0f, v = 0.0f;
      if (xl < W) {
        h = (float)__popc(s_lcen[xl] ^ rc[j]);
        u = fabsf(s_lcb[xl] - rcb4[j]);
        v = fabsf(s_lcr[xl] - rcr4[j]);
      }
      oh[j] = fmaf(h, AY, BY);
      ou[j] = fmaf(u, AU, BU);
      ov[j] = fmaf(v, AV, BV);
    }
    const size_t o = ob + (size_t)d * chs;
    __builtin_nontemporal_store(oh, (v4f*)(out + o));              // ham
    __builtin_nontemporal_store(ou, (v4f*)(out + o + 128 * chs));  // |dCb|
    __builtin_nontemporal_store(ov, (v4f*)(out + o + 256 * chs));  // |dCr|
  }
}

// ---------------- launch ----------------
extern "C" void kernel_launch(void* const* d_in, const int* in_sizes, int n_in,
                              void* d_out, int out_size, void* d_ws, size_t ws_size,
                              hipStream_t stream) {
  const float* left  = (const float*)d_in[0];
  const float* right = (const float*)d_in[1];
  float* out = (float*)d_out;

  // workspace layout: 8 planes of NHW (Y,CB,CR,CEN for left then right) = 7.9 MB
  float* LY   = (float*)d_ws;
  float* LCB  = LY  + NHW;
  float* LCR  = LCB + NHW;
  int*   LCEN = (int*)(LCR + NHW);
  float* RY   = (float*)(LCEN + NHW);
  float* RCB  = RY  + NHW;
  float* RCR  = RCB + NHW;
  int*   RCEN = (int*)(RCR + NHW);

  const dim3 g1(W / 128, H, NB), b1(64);
  fdscs_pool_ycbcr<<<g1, b1, 0, stream>>>(left,  LY, LCB, LCR);
  fdscs_pool_ycbcr<<<g1, b1, 0, stream>>>(right, RY, RCB, RCR);

  const dim3 g2(W / 128, H, NB), b2(128);
  fdscs_census<<<g2, b2, 0, stream>>>(LY, LCEN);
  fdscs_census<<<g2, b2, 0, stream>>>(RY, RCEN);

  const dim3 g3(H, NB, D / DCHUNK), b3(160);
  fdscs_costvol<<<g3, b3, 0, stream>>>(LCEN, LCB, LCR, RCEN, RCB, RCR, out);
}